// SoftMoERouter_14980845928886
// MI455X (gfx1250) — compile-verified
//
#include <hip/hip_runtime.h>
#include <hip/hip_bf16.h>
#include <math.h>

typedef __attribute__((ext_vector_type(2))) float v2f;
typedef __attribute__((ext_vector_type(8))) float v8f;

// Problem constants (match reference)
constexpr int T  = 16384;   // B*S tokens
constexpr int D  = 2048;
constexpr int NS = 64;      // num slots == E (SPE=1)
constexpr int E  = 64;
constexpr int H  = 1024;

// ---------------------------------------------------------------------------
// K1: logits[t,s] = sum_d x[t,d] * W_disp[s,d]       (T x NS x D f32 WMMA GEMM)
// One wave per 16x16 output tile. A = x tile (16 tok x 4 k), B = W_dispT tile.
// ---------------------------------------------------------------------------
__global__ __launch_bounds__(32) void k_logits(const float* __restrict__ x,
                                               const float* __restrict__ Wd,
                                               float* __restrict__ logits) {
  const int sn   = blockIdx.x * 16;       // slot tile base (0..48)
  const int tm   = blockIdx.y * 16;       // token tile base
  const int lane = threadIdx.x;
  const int mn   = lane & 15;             // M for A, N for B/C
  const int h    = lane >> 4;             // half-wave selects K pair / C rows
  const float* __restrict__ xrow = x  + (size_t)(tm + mn) * D;
  const float* __restrict__ wrow = Wd + (size_t)(sn + mn) * D;
  v8f c = {};
  for (int k = 0; k < D; k += 4) {
    v2f a = *(const v2f*)(xrow + k + 2 * h);   // A[m][k+2h], A[m][k+2h+1]
    v2f b = *(const v2f*)(wrow + k + 2 * h);   // B[k+2h][n] = Wd[n][k+2h]
    c = __builtin_amdgcn_wmma_f32_16x16x4_f32(false, a, false, b,
                                              (short)0, c, false, false);
  }
#pragma unroll
  for (int v = 0; v < 8; ++v) {
    const int row = tm + v + 8 * h;            // C: VGPR v -> M = v (+8 hi half)
    logits[(size_t)row * NS + sn + mn] = c[v];
  }
}

// ---------------------------------------------------------------------------
// K2: per-slot (column) max and sum(exp) over all T tokens, for dispatch softmax
// ---------------------------------------------------------------------------
__global__ __launch_bounds__(256) void k_colstats(const float* __restrict__ logits,
                                                  float* __restrict__ colmax,
                                                  float* __restrict__ colsum) {
  const int s   = blockIdx.x;                  // 0..63
  const int tid = threadIdx.x;
  __shared__ float red[256];
  float m = -3.4e38f;
  for (int t = tid; t < T; t += 256) m = fmaxf(m, logits[(size_t)t * NS + s]);
  red[tid] = m; __syncthreads();
  for (int w = 128; w > 0; w >>= 1) {
    if (tid < w) red[tid] = fmaxf(red[tid], red[tid + w]);
    __syncthreads();
  }
  m = red[0]; __syncthreads();
  float sum = 0.f;
  for (int t = tid; t < T; t += 256) sum += expf(logits[(size_t)t * NS + s] - m);
  red[tid] = sum; __syncthreads();
  for (int w = 128; w > 0; w >>= 1) {
    if (tid < w) red[tid] += red[tid + w];
    __syncthreads();
  }
  if (tid == 0) { colmax[s] = m; colsum[s] = red[0]; }
}

// ---------------------------------------------------------------------------
// K3: per-token row pass: combine = row-softmax(logits); dispatch = col-softmax
// One thread owns one token row of 64 logits (256B, cache-resident re-reads).
// ---------------------------------------------------------------------------
__global__ __launch_bounds__(256) void k_rowpass(const float* __restrict__ logits,
                                                 const float* __restrict__ colmax,
                                                 const float* __restrict__ colsum,
                                                 float* __restrict__ dispatchW,
                                                 float* __restrict__ combineW) {
  const int t = blockIdx.x * 256 + threadIdx.x;      // 0..T-1
  const float* __restrict__ lr = logits + (size_t)t * NS;
  float rmax = -3.4e38f;
  for (int i = 0; i < NS; ++i) rmax = fmaxf(rmax, lr[i]);
  float rsum = 0.f;
  for (int i = 0; i < NS; ++i) rsum += expf(lr[i] - rmax);
  const float rinv = 1.0f / rsum;
  for (int i = 0; i < NS; ++i) {
    const float li = lr[i];
    combineW [(size_t)t * NS + i] = expf(li - rmax) * rinv;
    dispatchW[(size_t)t * NS + i] = expf(li - colmax[i]) / colsum[i];
  }
}

// ---------------------------------------------------------------------------
// K4: slot_in[s,d] = sum_t dispatch[t,s] * x[t,d]   (NS x D x T f32 WMMA GEMM)
// Split-K by 16 over tokens, f32 atomic accumulation into workspace.
// ---------------------------------------------------------------------------
__global__ __launch_bounds__(32) void k_slotin(const float* __restrict__ dispatchW,
                                               const float* __restrict__ x,
                                               float* __restrict__ slot_in) {
  const int sm   = blockIdx.x * 16;       // slot tile
  const int dn   = blockIdx.y * 16;       // dim tile
  const int t0   = blockIdx.z * (T / 16); // split-K chunk of 1024 tokens
  const int lane = threadIdx.x;
  const int mn   = lane & 15;
  const int h    = lane >> 4;
  v8f c = {};
  for (int k = 0; k < T / 16; k += 4) {
    const int t = t0 + k + 2 * h;
    v2f a, b;
    a.x = dispatchW[(size_t)t       * NS + sm + mn];   // A[m][k]   = dispatchT[s][t]
    a.y = dispatchW[(size_t)(t + 1) * NS + sm + mn];
    b.x = x[(size_t)t       * D + dn + mn];            // B[k][n]   = x[t][d]
    b.y = x[(size_t)(t + 1) * D + dn + mn];
    c = __builtin_amdgcn_wmma_f32_16x16x4_f32(false, a, false, b,
                                              (short)0, c, false, false);
  }
#pragma unroll
  for (int v = 0; v < 8; ++v) {
    const int row = sm + v + 8 * h;
    atomicAdd(&slot_in[(size_t)row * D + dn + mn], c[v]);
  }
}

// ---------------------------------------------------------------------------
// K5: expert layer 1: h[e,j] = gelu( dot(W1[e,j,:], slot_in[e,:]) + b1[e,j] )
// Bandwidth-bound GEMV over 512 MiB of W1; slot_in row staged in LDS.
// ---------------------------------------------------------------------------
__global__ __launch_bounds__(128) void k_expert1(const float* __restrict__ slot_in,
                                                 const float* __restrict__ W1,
                                                 const float* __restrict__ b1,
                                                 float* __restrict__ hbuf) {
  const int e   = blockIdx.x;
  const int j   = blockIdx.y * 128 + threadIdx.x;     // hidden index
  __shared__ float xs[D];
  for (int i = threadIdx.x; i < D; i += 128) xs[i] = slot_in[(size_t)e * D + i];
  __syncthreads();
  const float* __restrict__ w = W1 + ((size_t)e * H + j) * D;
  float acc = 0.f;
  for (int k = 0; k < D; k += 4) {
    const float4 wv = *(const float4*)(w + k);
    acc += wv.x * xs[k] + wv.y * xs[k + 1] + wv.z * xs[k + 2] + wv.w * xs[k + 3];
  }
  acc += b1[(size_t)e * H + j];
  // exact (erf) GELU, matching torch F.gelu default
  hbuf[(size_t)e * H + j] = 0.5f * acc * (1.0f + erff(acc * 0.70710678118654752f));
}

// ---------------------------------------------------------------------------
// K6: expert layer 2: slot_out[e,d] = dot(W2[e,d,:], h[e,:]) + b2[e,d]
// ---------------------------------------------------------------------------
__global__ __launch_bounds__(128) void k_expert2(const float* __restrict__ hbuf,
                                                 const float* __restrict__ W2,
                                                 const float* __restrict__ b2,
                                                 float* __restrict__ slot_out) {
  const int e   = blockIdx.x;
  const int d   = blockIdx.y * 128 + threadIdx.x;
  __shared__ float hs[H];
  for (int i = threadIdx.x; i < H; i += 128) hs[i] = hbuf[(size_t)e * H + i];
  __syncthreads();
  const float* __restrict__ w = W2 + ((size_t)e * D + d) * H;
  float acc = 0.f;
  for (int k = 0; k < H; k += 4) {
    const float4 wv = *(const float4*)(w + k);
    acc += wv.x * hs[k] + wv.y * hs[k + 1] + wv.z * hs[k + 2] + wv.w * hs[k + 3];
  }
  slot_out[(size_t)e * D + d] = acc + b2[(size_t)e * D + d];
}

// ---------------------------------------------------------------------------
// K7: y[t,d] = sum_s combine[t,s] * slot_out[s,d]   (T x D x 64 f32 WMMA GEMM)
// Write-bandwidth bound (128 MiB out); combine/slot_out are L2-resident.
// ---------------------------------------------------------------------------
__global__ __launch_bounds__(32) void k_y(const float* __restrict__ combineW,
                                          const float* __restrict__ slot_out,
                                          float* __restrict__ y) {
  const int dn   = blockIdx.x * 16;
  const int tm   = blockIdx.y * 16;
  const int lane = threadIdx.x;
  const int mn   = lane & 15;
  const int h    = lane >> 4;
  const float* __restrict__ crow = combineW + (size_t)(tm + mn) * NS;
  v8f c = {};
  for (int k = 0; k < NS; k += 4) {
    v2f a = *(const v2f*)(crow + k + 2 * h);           // A[m][k] = combine[t][s]
    v2f b;
    b.x = slot_out[(size_t)(k + 2 * h)     * D + dn + mn];
    b.y = slot_out[(size_t)(k + 2 * h + 1) * D + dn + mn];
    c = __builtin_amdgcn_wmma_f32_16x16x4_f32(false, a, false, b,
                                              (short)0, c, false, false);
  }
#pragma unroll
  for (int v = 0; v < 8; ++v) {
    const int row = tm + v + 8 * h;
    y[(size_t)row * D + dn + mn] = c[v];
  }
}

// ---------------------------------------------------------------------------
extern "C" void kernel_launch(void* const* d_in, const int* in_sizes, int n_in,
                              void* d_out, int out_size, void* d_ws, size_t ws_size,
                              hipStream_t stream) {
  (void)in_sizes; (void)n_in; (void)out_size; (void)ws_size;
  const float* x   = (const float*)d_in[0];   // (T, D)
  const float* Wd  = (const float*)d_in[1];   // (NS, D)
  const float* W1  = (const float*)d_in[2];   // (E, H, D)
  const float* b1  = (const float*)d_in[3];   // (E, H)
  const float* W2  = (const float*)d_in[4];   // (E, D, H)
  const float* b2  = (const float*)d_in[5];   // (E, D)
  float* y = (float*)d_out;                   // (T, D)

  // workspace layout (f32 elements)
  float* ws        = (float*)d_ws;
  float* logits    = ws;                          // T*NS      = 1048576
  float* dispatchW = logits    + (size_t)T * NS;  // T*NS
  float* combineW  = dispatchW + (size_t)T * NS;  // T*NS
  float* colmax    = combineW  + (size_t)T * NS;  // 64
  float* colsum    = colmax + NS;                 // 64
  float* slot_in   = colsum + NS;                 // NS*D      = 131072
  float* hbuf      = slot_in + (size_t)NS * D;    // E*H       = 65536
  float* slot_out  = hbuf    + (size_t)E * H;     // E*D       = 131072

  // 1) logits = x @ Wd^T
  k_logits<<<dim3(NS / 16, T / 16), 32, 0, stream>>>(x, Wd, logits);
  // 2) per-slot softmax stats over tokens
  k_colstats<<<NS, 256, 0, stream>>>(logits, colmax, colsum);
  // 3) combine (row softmax) + dispatch (col softmax) weight matrices
  k_rowpass<<<T / 256, 256, 0, stream>>>(logits, colmax, colsum, dispatchW, combineW);
  // 4) slot_in = dispatch^T @ x  (split-K + f32 atomics, needs zeroed acc)
  hipMemsetAsync(slot_in, 0, (size_t)NS * D * sizeof(float), stream);
  k_slotin<<<dim3(NS / 16, D / 16, 16), 32, 0, stream>>>(dispatchW, x, slot_in);
  // 5/6) per-expert MLP with exact-erf GELU
  k_expert1<<<dim3(E, H / 128), 128, 0, stream>>>(slot_in, W1, b1, hbuf);
  k_expert2<<<dim3(E, D / 128), 128, 0, stream>>>(hbuf, W2, b2, slot_out);
  // 7) y = combine @ slot_out
  k_y<<<dim3(D / 16, T / 16), 32, 0, stream>>>(combineW, slot_out, y);
}